// Sinkhorn_seq_45311904973499
// MI455X (gfx1250) — compile-verified
//
#include <hip/hip_runtime.h>
#include <cstdint>
#include <cstddef>

#define B_ 8
#define S_ 128
#define V_ 4096

typedef __attribute__((ext_vector_type(2))) float v2f;
typedef __attribute__((ext_vector_type(4))) float f4;
typedef __attribute__((ext_vector_type(8))) float v8f;
typedef __attribute__((ext_vector_type(4))) int   v4i;

// ---------------- async global->LDS support (gfx1250) ----------------
#if defined(__HIP_DEVICE_COMPILE__) && defined(__gfx1250__) && \
    __has_builtin(__builtin_amdgcn_global_load_async_to_lds_b128)
#define HAVE_ASYNC 1
#else
#define HAVE_ASYNC 0
#endif

#if HAVE_ASYNC
typedef __attribute__((address_space(1))) v4i* gp128_t;
typedef __attribute__((address_space(3))) v4i* lp128_t;
#endif

__device__ __forceinline__ void wait_async_zero() {
#if HAVE_ASYNC
#if __has_builtin(__builtin_amdgcn_s_wait_asynccnt)
  __builtin_amdgcn_s_wait_asynccnt(0);
#else
  asm volatile("s_wait_asynccnt 0" ::: "memory");
#endif
#endif
}

// =====================================================================
// Kernel 1: row softmax  p = softmax(y / 2) over V=4096, one block/row
// =====================================================================
__global__ void sk_softmax(const float* __restrict__ ys,
                           const float* __restrict__ yt,
                           float* __restrict__ ps,
                           float* __restrict__ pt) {
  __shared__ float red[256];
  const int tid = threadIdx.x;
  const int row = blockIdx.x;                 // 0 .. 2*B*S-1
  const float* src;
  float* dst;
  if (row < B_ * S_) {
    src = ys + (size_t)row * V_;
    dst = ps + (size_t)row * V_;
  } else {
    const int r = row - B_ * S_;
    src = yt + (size_t)r * V_;
    dst = pt + (size_t)r * V_;
  }

  float m = -3.4e38f;
  for (int k = tid; k < V_; k += 256) m = fmaxf(m, src[k]);
  red[tid] = m;
  __syncthreads();
  for (int s = 128; s > 0; s >>= 1) {
    if (tid < s) red[tid] = fmaxf(red[tid], red[tid + s]);
    __syncthreads();
  }
  m = red[0];
  __syncthreads();

  float sum = 0.f;
  for (int k = tid; k < V_; k += 256) {
    const float e = __expf((src[k] - m) * 0.5f);  // softmax(y/2)
    dst[k] = e;
    sum += e;
  }
  red[tid] = sum;
  __syncthreads();
  for (int s = 128; s > 0; s >>= 1) {
    if (tid < s) red[tid] += red[tid + s];
    __syncthreads();
  }
  const float inv = 1.0f / red[0];
  for (int k = tid; k < V_; k += 256) dst[k] *= inv;
}

// =====================================================================
// Kernel 2: W[b][i][j] = sum_v |ps[b,i,v] - pt[b,j,v]|
// 32x32 tile per block, K staged through LDS in chunks of 512 floats.
// Uses global_load_async_to_lds_b128 when available.
// =====================================================================
#define KC 512
#define LDSTR (KC + 4)   // pad: 516 floats -> rows land on distinct banks

__global__ void sk_l1cdist(const float* __restrict__ ps,
                           const float* __restrict__ pt,
                           float* __restrict__ W) {
  __shared__ float Xs[32 * LDSTR];
  __shared__ float Ys[32 * LDSTR];
  const int tid = threadIdx.x;
  const int b  = blockIdx.y;
  const int ti = blockIdx.x >> 2;   // 0..3
  const int tj = blockIdx.x & 3;    // 0..3
  const float* xbase = ps + ((size_t)b * S_ + ti * 32) * V_;
  const float* ybase = pt + ((size_t)b * S_ + tj * 32) * V_;

  const int i0 = tid >> 4;   // 0..15
  const int j0 = tid & 15;   // 0..15
  float a00 = 0.f, a01 = 0.f, a10 = 0.f, a11 = 0.f;

  for (int kc = 0; kc < V_; kc += KC) {
    // stage 32 rows x 512 floats of each operand into LDS
    for (int idx = tid; idx < 32 * (KC / 4); idx += 256) {
      const int r  = idx >> 7;            // row 0..31
      const int c4 = (idx & 127) << 2;    // float offset within chunk
      const float* gx = xbase + (size_t)r * V_ + kc + c4;
      const float* gy = ybase + (size_t)r * V_ + kc + c4;
#if HAVE_ASYNC
      __builtin_amdgcn_global_load_async_to_lds_b128(
          (gp128_t)const_cast<float*>(gx),
          (lp128_t)&Xs[r * LDSTR + c4], 0, 0);
      __builtin_amdgcn_global_load_async_to_lds_b128(
          (gp128_t)const_cast<float*>(gy),
          (lp128_t)&Ys[r * LDSTR + c4], 0, 0);
#else
      *(f4*)&Xs[r * LDSTR + c4] = *(const f4*)gx;
      *(f4*)&Ys[r * LDSTR + c4] = *(const f4*)gy;
#endif
    }
    wait_async_zero();
    __syncthreads();

#pragma unroll 4
    for (int v = 0; v < KC; v += 4) {
      const f4 xa = *(const f4*)&Xs[i0 * LDSTR + v];
      const f4 xb = *(const f4*)&Xs[(i0 + 16) * LDSTR + v];
      const f4 ya = *(const f4*)&Ys[j0 * LDSTR + v];
      const f4 yb = *(const f4*)&Ys[(j0 + 16) * LDSTR + v];
      a00 += fabsf(xa.x - ya.x) + fabsf(xa.y - ya.y) +
             fabsf(xa.z - ya.z) + fabsf(xa.w - ya.w);
      a01 += fabsf(xa.x - yb.x) + fabsf(xa.y - yb.y) +
             fabsf(xa.z - yb.z) + fabsf(xa.w - yb.w);
      a10 += fabsf(xb.x - ya.x) + fabsf(xb.y - ya.y) +
             fabsf(xb.z - ya.z) + fabsf(xb.w - ya.w);
      a11 += fabsf(xb.x - yb.x) + fabsf(xb.y - yb.y) +
             fabsf(xb.z - yb.z) + fabsf(xb.w - yb.w);
    }
    __syncthreads();
  }

  float* wrow = W + (size_t)b * S_ * S_;
  wrow[(ti * 32 + i0)      * S_ + tj * 32 + j0]      = a00;
  wrow[(ti * 32 + i0)      * S_ + tj * 32 + j0 + 16] = a01;
  wrow[(ti * 32 + i0 + 16) * S_ + tj * 32 + j0]      = a10;
  wrow[(ti * 32 + i0 + 16) * S_ + tj * 32 + j0 + 16] = a11;
}

// =====================================================================
// Kernel 3: per-batch Sinkhorn. P kept in LDS (128x129 padded).
// Row/col sums computed with V_WMMA_F32_16X16X4_F32 (B = ones, so only
// the documented A and C/D lane layouts matter).
// One block of 256 threads (8 waves) per batch; each wave owns a
// 16-row (or 16-col) stripe and chains 32 WMMAs to reduce K=128.
// =====================================================================
__global__ void sk_sinkhorn(const float* __restrict__ W,
                            float* __restrict__ partial) {
  __shared__ float Pl[S_ * 129];
  __shared__ float rs[S_];
  __shared__ float inv[S_];
  __shared__ float red[256];

  const int tid = threadIdx.x;
  const int b = blockIdx.x;
  const float* Wg = W + (size_t)b * S_ * S_;

  // P = exp(-W / 0.1)
  for (int idx = tid; idx < S_ * S_; idx += 256) {
    const int i = idx >> 7, j = idx & 127;
    Pl[i * 129 + j] = __expf(-10.0f * Wg[idx]);
  }
  __syncthreads();

  const int wave = tid >> 5;      // 0..7
  const int lane = tid & 31;
  const int lm   = lane & 15;     // M index for A layout
  const int h    = lane >> 4;     // half: K pair selector
  v2f ones;
  ones.x = 1.0f;
  ones.y = 1.0f;

  for (int it = 0; it < 10; ++it) {
    // ---- row sums: D = P_stripe(16x4 chunks) x ones ----
    {
      v8f d = {};
      const int rowbase = (wave * 16 + lm) * 129;
      const int koff = 2 * h;
#pragma unroll
      for (int c = 0; c < 32; ++c) {
        v2f a;
        a.x = Pl[rowbase + 4 * c + koff];
        a.y = Pl[rowbase + 4 * c + koff + 1];
        d = __builtin_amdgcn_wmma_f32_16x16x4_f32(
            false, a, false, ones, (short)0, d, false, false);
      }
      if (lane == 0 || lane == 16) {
        const int rbase = wave * 16 + h * 8;   // lanes 0..15 -> M=r, 16..31 -> M=8+r
#pragma unroll
        for (int r = 0; r < 8; ++r) rs[rbase + r] = d[r];
      }
    }
    __syncthreads();
    if (tid < S_) inv[tid] = 1.0f / rs[tid];
    __syncthreads();
    for (int idx = tid; idx < S_ * S_; idx += 256) {
      const int i = idx >> 7, j = idx & 127;
      Pl[i * 129 + j] *= inv[i];
    }
    __syncthreads();

    // ---- col sums: load A transposed from LDS, same WMMA ----
    {
      v8f d = {};
      const int colbase = wave * 16 + lm;
#pragma unroll
      for (int c = 0; c < 32; ++c) {
        const int r0 = 4 * c + 2 * h;
        v2f a;
        a.x = Pl[r0 * 129 + colbase];
        a.y = Pl[(r0 + 1) * 129 + colbase];
        d = __builtin_amdgcn_wmma_f32_16x16x4_f32(
            false, a, false, ones, (short)0, d, false, false);
      }
      if (lane == 0 || lane == 16) {
        const int cbase = wave * 16 + h * 8;
#pragma unroll
        for (int r = 0; r < 8; ++r) rs[cbase + r] = d[r];
      }
    }
    __syncthreads();
    if (tid < S_) inv[tid] = 1.0f / rs[tid];
    __syncthreads();
    for (int idx = tid; idx < S_ * S_; idx += 256) {
      const int i = idx >> 7, j = idx & 127;
      Pl[i * 129 + j] *= inv[j];
    }
    __syncthreads();
  }

  // sum(P * W)
  float acc = 0.f;
  for (int idx = tid; idx < S_ * S_; idx += 256) {
    const int i = idx >> 7, j = idx & 127;
    acc += Pl[i * 129 + j] * Wg[idx];
  }
  red[tid] = acc;
  __syncthreads();
  for (int s = 128; s > 0; s >>= 1) {
    if (tid < s) red[tid] += red[tid + s];
    __syncthreads();
  }
  if (tid == 0) partial[b] = red[0];
}

// =====================================================================
// Kernel 4: deterministic finalize (fixed-order sum of 8 partials)
// =====================================================================
__global__ void sk_finalize(const float* __restrict__ partial,
                            float* __restrict__ out) {
  if (threadIdx.x == 0) {
    float s = 0.f;
    for (int b = 0; b < B_; ++b) s += partial[b];
    out[0] = 0.001f * s;
  }
}

// =====================================================================
extern "C" void kernel_launch(void* const* d_in, const int* in_sizes, int n_in,
                              void* d_out, int out_size, void* d_ws, size_t ws_size,
                              hipStream_t stream) {
  (void)in_sizes; (void)n_in; (void)out_size; (void)ws_size;
  const float* ys = (const float*)d_in[0];
  const float* yt = (const float*)d_in[1];
  float* ws = (float*)d_ws;

  float* ps      = ws;                                  // B*S*V
  float* pt      = ps + (size_t)B_ * S_ * V_;           // B*S*V
  float* W       = pt + (size_t)B_ * S_ * V_;           // B*S*S
  float* partial = W + (size_t)B_ * S_ * S_;            // B
  float* out     = (float*)d_out;

  sk_softmax <<<2 * B_ * S_, 256, 0, stream>>>(ys, yt, ps, pt);
  sk_l1cdist <<<dim3(16, B_), 256, 0, stream>>>(ps, pt, W);
  sk_sinkhorn<<<B_, 256, 0, stream>>>(W, partial);
  sk_finalize<<<1, 32, 0, stream>>>(partial, out);
}